// RandomFlipLayer_88493506166927
// MI455X (gfx1250) — compile-verified
//
#include <hip/hip_runtime.h>

// RandomFlipLayer: out[b,h,w,c] = x[b, row_perm[b,h], col_perm[b,w], p_b[c]]
// B=256, H=128, W=128, C=9, float32. Pure gather -> HBM-bandwidth bound
// (~302 MB total traffic => ~13 us floor at 23.3 TB/s).
//
// Strategy: 1 block per output row (b,h). 288 threads = 9 wave32s, one float4
// per thread covers the 1152-float contiguous source row. Stage the row into
// LDS via CDNA5 async global->LDS B128 copies (ASYNCcnt), then emit fully
// coalesced float4 stores whose components are gathered from LDS.

typedef int v4i __attribute__((ext_vector_type(4)));
typedef __attribute__((address_space(1))) v4i* global_v4i_ptr;
typedef __attribute__((address_space(3))) v4i* lds_v4i_ptr;

// Channel permutations packed as 9 x 4-bit entries (entry c at bits [4c+3:4c]).
// P1 = [0,3,4,1,2,5,7,6,8], P2 = [0,1,2,4,3,7,8,5,6], P3 = [0,2,1,3,4,6,5,8,7]
static constexpr unsigned long long P1PACK = 0x867521430ull;
static constexpr unsigned long long P2PACK = 0x658734210ull;
static constexpr unsigned long long P3PACK = 0x785643120ull;

static __device__ __forceinline__ unsigned permStep(unsigned long long pack, unsigned c) {
    return (unsigned)((pack >> (c * 4u)) & 0xFull);
}

__global__ __launch_bounds__(288) void RandomFlipLayer_88493506166927_kernel(
    const float* __restrict__ x,
    const int*  __restrict__ row_perm,
    const int*  __restrict__ col_perm,
    const int*  __restrict__ flags,
    float*      __restrict__ out)
{
    constexpr int H = 128, W = 128, C = 9;
    constexpr int ROWF = W * C;              // 1152 floats = 4608 bytes per row

    __shared__ float row_lds[ROWF];          // staged source row
    __shared__ int   cp_lds[W];              // staged per-sample column perm

    const int blk = blockIdx.x;              // 0 .. B*H-1
    const int b   = blk >> 7;                // blk / H
    const int h   = blk & (H - 1);           // blk % H
    const int tid = threadIdx.x;             // 0 .. 287

    // Uniform per-block scalars (b,h uniform -> scalar-cache friendly).
    const int rp = row_perm[b * H + h];
    const int f0 = flags[b * 3 + 0];
    const int f1 = flags[b * 3 + 1];
    const int f2 = flags[b * 3 + 2];

    // Stage column permutation row into LDS.
    if (tid < W) cp_lds[tid] = col_perm[b * W + tid];

    // Stage the (row-permuted) source row into LDS: 288 lanes x 16B = 4608B.
    {
        const float* g = x + ((size_t)b * H + (size_t)rp) * ROWF + tid * 4;
        float*       l = &row_lds[tid * 4];
#if __has_builtin(__builtin_amdgcn_global_load_async_to_lds_b128)
        __builtin_amdgcn_global_load_async_to_lds_b128(
            (global_v4i_ptr)(float*)g,   // global src (cast away const, then addrspace)
            (lds_v4i_ptr)l,              // LDS dst
            /*imm offset=*/0, /*cpol=*/0);
  #if __has_builtin(__builtin_amdgcn_s_wait_asynccnt)
        __builtin_amdgcn_s_wait_asynccnt(0);
  #else
        asm volatile("s_wait_asynccnt 0" ::: "memory");
  #endif
#else
        *(float4*)l = *(const float4*)g;
#endif
    }
    __syncthreads();

    // Each thread produces one coalesced float4 of the output row.
    const int base = tid * 4;                // flat index into the 1152-float row
    float4 v;
    float* vp = &v.x;
#pragma unroll
    for (int k = 0; k < 4; ++k) {
        const int j = base + k;
        const int w = j / 9;                 // pixel
        const int c = j - w * 9;             // channel
        // Compose channel permutation (reference order: apply P3, then P2, then P1).
        unsigned cc = (unsigned)c;
        if (f2 > 0) cc = permStep(P3PACK, cc);
        if (f1 > 0) cc = permStep(P2PACK, cc);
        if (f0 > 0) cc = permStep(P1PACK, cc);
        vp[k] = row_lds[cp_lds[w] * 9 + (int)cc];
    }

    float* dst = out + ((size_t)b * H + (size_t)h) * ROWF + base;
    *(float4*)dst = v;
}

extern "C" void kernel_launch(void* const* d_in, const int* in_sizes, int n_in,
                              void* d_out, int out_size, void* d_ws, size_t ws_size,
                              hipStream_t stream) {
    (void)in_sizes; (void)n_in; (void)out_size; (void)d_ws; (void)ws_size;
    const float* x        = (const float*)d_in[0];
    const int*   row_perm = (const int*)d_in[1];
    const int*   col_perm = (const int*)d_in[2];
    const int*   flags    = (const int*)d_in[3];
    float*       out      = (float*)d_out;

    constexpr int B = 256, H = 128;
    dim3 grid(B * H);     // one block per output row
    dim3 block(288);      // 9 wave32s; one float4 per thread per phase
    RandomFlipLayer_88493506166927_kernel<<<grid, block, 0, stream>>>(
        x, row_perm, col_perm, flags, out);
}